// GLAttentionGeneral_21869973471395
// MI455X (gfx1250) — compile-verified
//
#include <hip/hip_runtime.h>
#include <math.h>

#define NB   16
#define IDF  128
#define CDF  256
#define Q_   16384
#define SL   48

typedef __attribute__((ext_vector_type(2))) float v2f;
typedef __attribute__((ext_vector_type(8))) float v8f;

__device__ __forceinline__ v8f wmma4(v2f a, v2f b, v8f c) {
  // D = A(16x4 f32) * B(4x16 f32) + C(16x16 f32), full f32 precision
  return __builtin_amdgcn_wmma_f32_16x16x4_f32(false, a, false, b, (short)0, c,
                                               false, false);
}

// ---------------------------------------------------------------------------
// Kernel 1: sourceT[b,i,s] = sum_c w_ctx[i,c]*context[b,c,s]
//           sent[b,i]      = b_lin[i] + sum_j sentence[b,j]*w_lin[i,j]
// ---------------------------------------------------------------------------
__global__ __launch_bounds__(256) void prep_kernel(
    const float* __restrict__ sentence, const float* __restrict__ context,
    const float* __restrict__ w_ctx, const float* __restrict__ w_lin,
    const float* __restrict__ b_lin, float* __restrict__ sourceT,
    float* __restrict__ sent) {
  extern __shared__ float smem[];           // context tile: 256*48 floats
  const int b = blockIdx.x;
  const int t = threadIdx.x;

  for (int idx = t; idx < CDF * SL; idx += 256)
    smem[idx] = context[(size_t)b * CDF * SL + idx];
  __syncthreads();

  const int i = t & 127;
  const int sbase = (t >> 7) * 24;          // two thread-groups cover 48 s
  float acc[24];
#pragma unroll
  for (int s2 = 0; s2 < 24; ++s2) acc[s2] = 0.0f;
  for (int c = 0; c < CDF; ++c) {
    const float w = w_ctx[i * CDF + c];
    const float* cr = &smem[c * SL + sbase];
#pragma unroll
    for (int s2 = 0; s2 < 24; ++s2) acc[s2] += w * cr[s2];
  }
#pragma unroll
  for (int s2 = 0; s2 < 24; ++s2)
    sourceT[((size_t)b * IDF + i) * SL + sbase + s2] = acc[s2];

  if (t < IDF) {
    float a = b_lin[t];
    for (int j = 0; j < 100; ++j) a += sentence[b * 100 + j] * w_lin[t * 100 + j];
    sent[b * IDF + t] = a;
  }
}

// ---------------------------------------------------------------------------
// Kernel 2: attention.  block = (q-tile of 128, batch), 8 waves x 16 q.
//   attn[q,s] = sum_i target[i,q]*sourceT[i,s]  (WMMA, M=q,N=s,K=i)
//   mask -> -inf, softmax over s, write word_attn, then
//   wc[i,q] = sum_s sourceT[i,s]*attn[q,s]      (WMMA, M=i,N=q,K=s)
// ---------------------------------------------------------------------------
__global__ __launch_bounds__(256) void attn_kernel(
    const float* __restrict__ input, const float* __restrict__ sourceT,
    const int* __restrict__ mask, float* __restrict__ wc,
    float* __restrict__ wattn) {
  extern __shared__ float smem[];
  float* sT = smem;                         // [128][49]  padded sourceT
  const int b = blockIdx.y;
  const int t = threadIdx.x;
  const int wave = t >> 5;
  const int lane = t & 31;
  const int ln = lane & 15;
  const int hi = lane >> 4;
  float* aTw = smem + IDF * 49 + wave * (SL * 17);   // [48][17] attn^T stage

  // stage sourceT[b] into LDS with pad-49 rows
  const float* sTg = sourceT + (size_t)b * IDF * SL;
  for (int idx = t; idx < IDF * SL; idx += 256)
    sT[(idx / SL) * 49 + (idx % SL)] = sTg[idx];
  __syncthreads();

  const int q0 = blockIdx.x * 128 + wave * 16;
  const float* tg = input + (size_t)b * IDF * Q_;

  // ---- GEMM1: D[q, s] tiles, N-tiles s0 = 0,16,32, K = 128 ----
  v8f acc0 = {}, acc1 = {}, acc2 = {};
  for (int k = 0; k < IDF; k += 4) {
    const int kk = k + hi * 2;
    const float* ap = tg + (size_t)kk * Q_ + q0 + ln;
    v2f a;  a.x = ap[0];  a.y = ap[Q_];                 // K and K+1
    const float* br = &sT[kk * 49];
    v2f b0; b0.x = br[ln];      b0.y = br[49 + ln];
    acc0 = wmma4(a, b0, acc0);
    v2f b1; b1.x = br[16 + ln]; b1.y = br[49 + 16 + ln];
    acc1 = wmma4(a, b1, acc1);
    v2f b2; b2.x = br[32 + ln]; b2.y = br[49 + 32 + ln];
    acc2 = wmma4(a, b2, acc2);
  }

  // ---- mask + softmax over s (rows: lanes 0-15 hold q=r, 16-31 q=r+8) ----
  const int* mrow = mask + b * SL;
  const int m0 = mrow[ln], m1 = mrow[16 + ln], m2 = mrow[32 + ln];
  const float NEG = -__builtin_inff();
#pragma unroll
  for (int r = 0; r < 8; ++r) {
    float v0 = m0 ? NEG : acc0[r];
    float v1 = m1 ? NEG : acc1[r];
    float v2 = m2 ? NEG : acc2[r];
    float mx = fmaxf(fmaxf(v0, v1), v2);
    mx = fmaxf(mx, __shfl_xor(mx, 1));
    mx = fmaxf(mx, __shfl_xor(mx, 2));
    mx = fmaxf(mx, __shfl_xor(mx, 4));
    mx = fmaxf(mx, __shfl_xor(mx, 8));
    float p0 = __expf(v0 - mx), p1 = __expf(v1 - mx), p2 = __expf(v2 - mx);
    float s = p0 + p1 + p2;
    s += __shfl_xor(s, 1); s += __shfl_xor(s, 2);
    s += __shfl_xor(s, 4); s += __shfl_xor(s, 8);
    const float inv = 1.0f / s;
    const int ql = r + hi * 8;                    // local q of this element
    aTw[(ln)      * 17 + ql] = p0 * inv;          // attn^T[s][q_local]
    aTw[(16 + ln) * 17 + ql] = p1 * inv;
    aTw[(32 + ln) * 17 + ql] = p2 * inv;
  }
  __syncthreads();

  // word_attn[b,s,q] = attn^T rows, coalesced 64B per half-wave
  for (int s = hi; s < SL; s += 2)
    wattn[((size_t)b * SL + s) * Q_ + q0 + ln] = aTw[s * 17 + ln];

  // ---- GEMM2: wc[i, q] = sourceT(128x48) x attn^T(48x16), K = 48 ----
  v2f Bv[12];
#pragma unroll
  for (int ks = 0; ks < 12; ++ks) {
    const int k0 = ks * 4;
    Bv[ks].x = aTw[(k0 + hi * 2) * 17 + ln];
    Bv[ks].y = aTw[(k0 + 1 + hi * 2) * 17 + ln];
  }
#pragma unroll
  for (int mt = 0; mt < 8; ++mt) {
    const int mrow0 = mt * 16;
    v8f acc = {};
#pragma unroll
    for (int ks = 0; ks < 12; ++ks) {
      const int k0 = ks * 4;
      v2f a;
      a.x = sT[(mrow0 + ln) * 49 + k0 + hi * 2];
      a.y = sT[(mrow0 + ln) * 49 + k0 + 1 + hi * 2];
      acc = wmma4(a, Bv[ks], acc);
    }
    float* wcb = wc + ((size_t)b * IDF + mrow0 + hi * 8) * Q_ + q0 + ln;
#pragma unroll
    for (int r = 0; r < 8; ++r) wcb[(size_t)r * Q_] = acc[r];
  }
}

// ---------------------------------------------------------------------------
// Kernel 3: u^T[q,o] = sum_i (input[b,i,q]*sent[b,i]) * w_sv[o,i]
//           softmax over o; sent_att and weightedSentence = sent[o]*sent_att
// ---------------------------------------------------------------------------
__global__ __launch_bounds__(256) void sent_kernel(
    const float* __restrict__ input, const float* __restrict__ w_sv,
    const float* __restrict__ sent, float* __restrict__ wsent,
    float* __restrict__ satt) {
  extern __shared__ float smem[];
  float* wv    = smem;                       // [128][129] w_sv padded
  float* stg   = smem + IDF * 129;           // 8 waves * [128][17]
  float* sentl = stg + 8 * IDF * 17;         // [128]
  const int b = blockIdx.y;
  const int t = threadIdx.x;
  const int wave = t >> 5;
  const int lane = t & 31;
  const int ln = lane & 15;
  const int hi = lane >> 4;

  for (int idx = t; idx < IDF * IDF; idx += 256)
    wv[(idx >> 7) * 129 + (idx & 127)] = w_sv[idx];
  if (t < IDF) sentl[t] = sent[b * IDF + t];
  __syncthreads();

  const int q0 = blockIdx.x * 128 + wave * 16;
  const float* inb = input + (size_t)b * IDF * Q_;

  v8f acc[8];
#pragma unroll
  for (int nt = 0; nt < 8; ++nt) acc[nt] = (v8f){};

  for (int k = 0; k < IDF; k += 4) {
    const int kk = k + hi * 2;
    v2f a;
    a.x = inb[(size_t)kk * Q_ + q0 + ln] * sentl[kk];
    a.y = inb[(size_t)(kk + 1) * Q_ + q0 + ln] * sentl[kk + 1];
#pragma unroll
    for (int nt = 0; nt < 8; ++nt) {
      v2f bb;
      bb.x = wv[(nt * 16 + ln) * 129 + kk];      // w_sv[o, K]
      bb.y = wv[(nt * 16 + ln) * 129 + kk + 1];  // w_sv[o, K+1]
      acc[nt] = wmma4(a, bb, acc[nt]);
    }
  }

  // softmax over o (8 N-tiles x 16 lanes within each half), stage transposed
  float* stgw = stg + wave * (IDF * 17);
#pragma unroll
  for (int r = 0; r < 8; ++r) {
    float mx = acc[0][r];
#pragma unroll
    for (int nt = 1; nt < 8; ++nt) mx = fmaxf(mx, acc[nt][r]);
    mx = fmaxf(mx, __shfl_xor(mx, 1));
    mx = fmaxf(mx, __shfl_xor(mx, 2));
    mx = fmaxf(mx, __shfl_xor(mx, 4));
    mx = fmaxf(mx, __shfl_xor(mx, 8));
    float s = 0.0f;
    float p[8];
#pragma unroll
    for (int nt = 0; nt < 8; ++nt) { p[nt] = __expf(acc[nt][r] - mx); s += p[nt]; }
    s += __shfl_xor(s, 1); s += __shfl_xor(s, 2);
    s += __shfl_xor(s, 4); s += __shfl_xor(s, 8);
    const float inv = 1.0f / s;
    const int ql = r + hi * 8;
#pragma unroll
    for (int nt = 0; nt < 8; ++nt)
      stgw[(nt * 16 + ln) * 17 + ql] = p[nt] * inv;   // stage[o][q_local]
  }
  __syncthreads();

  // coalesced stores of both outputs from staged [o][q] tile
  const size_t ob = (size_t)b * IDF * Q_ + q0 + ln;
  for (int o = hi; o < IDF; o += 2) {
    const float v = stgw[o * 17 + ln];
    const size_t off = ob + (size_t)o * Q_;
    satt[off]  = v;
    wsent[off] = sentl[o] * v;
  }
}

// ---------------------------------------------------------------------------
extern "C" void kernel_launch(void* const* d_in, const int* in_sizes, int n_in,
                              void* d_out, int out_size, void* d_ws, size_t ws_size,
                              hipStream_t stream) {
  const float* input    = (const float*)d_in[0];
  const float* sentence = (const float*)d_in[1];
  const float* context  = (const float*)d_in[2];
  const int*   mask     = (const int*)d_in[3];
  const float* w_ctx    = (const float*)d_in[4];
  const float* w_sv     = (const float*)d_in[5];
  const float* w_lin    = (const float*)d_in[6];
  const float* b_lin    = (const float*)d_in[7];

  float* out = (float*)d_out;
  float* wc    = out;                                   // (16,128,128,128)
  float* wsent = out + (size_t)NB * IDF * Q_;           // (16,128,128,128)
  float* wattn = wsent + (size_t)NB * IDF * Q_;         // (16,48,128,128)
  float* satt  = wattn + (size_t)NB * SL * Q_;          // (16,128,128,128)

  float* ws_sourceT = (float*)d_ws;                     // 16*128*48 floats
  float* ws_sent    = ws_sourceT + (size_t)NB * IDF * SL;

  prep_kernel<<<NB, 256, CDF * SL * sizeof(float), stream>>>(
      sentence, context, w_ctx, w_lin, b_lin, ws_sourceT, ws_sent);

  dim3 grid(Q_ / 128, NB);
  const size_t attn_lds = (IDF * 49 + 8 * SL * 17) * sizeof(float);   // ~51 KB
  attn_kernel<<<grid, 256, attn_lds, stream>>>(input, ws_sourceT, mask, wc, wattn);

  const size_t sent_lds = (IDF * 129 + 8 * IDF * 17 + IDF) * sizeof(float); // ~136 KB
  sent_kernel<<<grid, 256, sent_lds, stream>>>(input, w_sv, ws_sent, wsent, satt);
}